// RotationLayer_82970178224742
// MI455X (gfx1250) — compile-verified
//
#include <hip/hip_runtime.h>

// Native ext-vector types so __builtin_nontemporal_load accepts them and we
// get single b128 / b64 global loads.
typedef int   v4i __attribute__((ext_vector_type(4)));
typedef float v2f __attribute__((ext_vector_type(2)));
typedef float v4f __attribute__((ext_vector_type(4)));

// Lower float atomicAdd to hardware global_atomic_add_f32 (no CAS loop).
#if defined(__has_cpp_attribute)
#if __has_cpp_attribute(clang::atomic)
#define FP_ATOMIC_HINT [[clang::atomic(no_remote_memory, no_fine_grained_memory, ignore_denormal_mode)]]
#else
#define FP_ATOMIC_HINT
#endif
#else
#define FP_ATOMIC_HINT
#endif

// ---------------------------------------------------------------------------
// Per-edge rotate + scatter-add, TWO edges per thread for ILP:
//   t_u = R(+phase) * x[u]  accumulated into out[v]
//   t_v = R(-phase) * x[v]  accumulated into out[u]
// Streamed-once data (edges 256 MB + phases 64 MB) uses non-temporal loads so
// it doesn't evict the 16 MB hot set (x + out) from the 192 MB L2.
// ---------------------------------------------------------------------------
__global__ __launch_bounds__(256) void rot_scatter_kernel(
    const v2f* __restrict__ x,
    const float* __restrict__ phases,
    const v4i* __restrict__ edges,   // one int4 = {u_lo, u_hi, v_lo, v_hi}
    float* __restrict__ out,
    int n_edges) {
    int t  = blockIdx.x * blockDim.x + threadIdx.x;
    int e0 = 2 * t;
    if (e0 >= n_edges) return;
    bool has_e1 = (e0 + 1) < n_edges;

    // Independent streamed loads: 1x b64 (two phases) + 2x b128 (two edges).
    v2f pp  = __builtin_nontemporal_load((const v2f*)(phases + e0));
    v4i ed0 = __builtin_nontemporal_load(edges + e0);
    v4i ed1 = __builtin_nontemporal_load(edges + (has_e1 ? e0 + 1 : e0));

    int u0 = ed0.x, v0 = ed0.z;      // node ids < 2^20, high dwords zero
    int u1 = ed1.x, v1 = ed1.z;

    float s0, c0, s1, c1;
    __sincosf(pp.x, &s0, &c0);       // v_sin_f32 / v_cos_f32 fast path
    __sincosf(pp.y, &s1, &c1);

    // Hot, L2-resident gathers (RT temporal hint).
    v2f hu0 = x[u0];
    v2f hv0 = x[v0];
    v2f hu1 = x[u1];
    v2f hv1 = x[v1];

    float tux0 = hu0.x * c0 - hu0.y * s0;   // R(+p0) * h_u0
    float tuy0 = hu0.x * s0 + hu0.y * c0;
    float tvx0 = hv0.x * c0 + hv0.y * s0;   // R(-p0) * h_v0
    float tvy0 = hv0.y * c0 - hv0.x * s0;

    float tux1 = hu1.x * c1 - hu1.y * s1;   // R(+p1) * h_u1
    float tuy1 = hu1.x * s1 + hu1.y * c1;
    float tvx1 = hv1.x * c1 + hv1.y * s1;   // R(-p1) * h_v1
    float tvy1 = hv1.y * c1 - hv1.x * s1;

    FP_ATOMIC_HINT {
        atomicAdd(out + 2 * v0 + 0, tux0);
        atomicAdd(out + 2 * v0 + 1, tuy0);
        atomicAdd(out + 2 * u0 + 0, tvx0);
        atomicAdd(out + 2 * u0 + 1, tvy0);
        if (has_e1) {
            atomicAdd(out + 2 * v1 + 0, tux1);
            atomicAdd(out + 2 * v1 + 1, tuy1);
            atomicAdd(out + 2 * u1 + 0, tvx1);
            atomicAdd(out + 2 * u1 + 1, tvy1);
        }
    }
}

// ---------------------------------------------------------------------------
// out = x  (seed the accumulator; harness poisons d_out)
// 2,000,000 floats -> 500,000 float4 stores, fully coalesced.
// ---------------------------------------------------------------------------
__global__ __launch_bounds__(256) void rot_init_kernel(const v4f* __restrict__ x4,
                                                       v4f* __restrict__ out4,
                                                       int n4) {
    int i = blockIdx.x * blockDim.x + threadIdx.x;
    if (i < n4) {
        out4[i] = x4[i];
    }
}

// ---------------------------------------------------------------------------
// Launch
// Inputs (setup_inputs order): d_in[0]=x (float32, 2M elems),
//                              d_in[1]=phases (float32, 16M elems),
//                              d_in[2]=edges (int64, 32M elems = 16M edges)
// d_out: 2M float32.
// ---------------------------------------------------------------------------
extern "C" void kernel_launch(void* const* d_in, const int* in_sizes, int n_in,
                              void* d_out, int out_size, void* d_ws, size_t ws_size,
                              hipStream_t stream) {
    const v2f*   x      = (const v2f*)d_in[0];
    const float* phases = (const float*)d_in[1];
    const v4i*   edges  = (const v4i*)d_in[2];
    float*       out    = (float*)d_out;

    const int n_node_floats = in_sizes[0];      // 2,000,000
    const int n_edges       = in_sizes[1];      // 16,000,000

    // Seed out = x (must run before the scatter; same stream => ordered).
    const int n4 = n_node_floats / 4;
    rot_init_kernel<<<(n4 + 255) / 256, 256, 0, stream>>>(
        (const v4f*)d_in[0], (v4f*)d_out, n4);

    // Edge scatter: two edges per thread.
    const int threads   = 256;
    const int n_threads = (n_edges + 1) / 2;
    const int blocks    = (n_threads + threads - 1) / threads;
    rot_scatter_kernel<<<blocks, threads, 0, stream>>>(x, phases, edges, out, n_edges);
}